// CorrelationLayer_14851996910164
// MI455X (gfx1250) — compile-verified
//
#include <hip/hip_runtime.h>

typedef __attribute__((ext_vector_type(2))) float v2f;
typedef __attribute__((ext_vector_type(8))) float v8f;

#define B_   8
#define C_   256
#define H_   96
#define W_   96
#define HW_  (H_ * W_)          // 9216

// ---------------------------------------------------------------------------
// Kernel 1: m2[b,c] = mean over HW of fmap2[b,c,:,:]
// One 256-thread block per (b,c) row; 9216 floats = 2304 float4 -> 9 per lane.
// Streams fmap2 (75.5 MB) once with b128 loads; 16384 waves -> latency covered.
// ---------------------------------------------------------------------------
__global__ __launch_bounds__(256) void corr_mean_kernel(
    const float* __restrict__ fmap2, float* __restrict__ m2) {
  const int row = blockIdx.x;                      // b*C + c, 0..2047
  const float4* p4 = (const float4*)(fmap2 + (size_t)row * HW_);
  float s = 0.f;
#pragma unroll
  for (int i = 0; i < 9; ++i) {
    float4 v = p4[threadIdx.x + i * 256];
    s += (v.x + v.y) + (v.z + v.w);
  }
  // wave32 butterfly reduce
#pragma unroll
  for (int off = 16; off > 0; off >>= 1)
    s += __shfl_xor(s, off, 32);
  __shared__ float ws[8];
  const int wv   = threadIdx.x >> 5;
  const int lane = threadIdx.x & 31;
  if (lane == 0) ws[wv] = s;
  __syncthreads();
  if (threadIdx.x == 0) {
    float t = 0.f;
#pragma unroll
    for (int w = 0; w < 8; ++w) t += ws[w];
    m2[row] = t * (1.0f / (float)HW_);
  }
}

// ---------------------------------------------------------------------------
// Kernel 2: corr[b,hw] = sum_c fmap1[b,c,hw] * m2[b,c] via V_WMMA_F32_16X16X4_F32
//   A(16x4)  = fmap1 tile: M = hw within 16-tile, K = channel (4 per step)
//   B(4x16)  = m2[c0..c0+3] broadcast across all 16 N columns
//   D(16x16) = every N column identical == corr tile, f32 accumulation
// Split-K: wave pairs (w, w+4) each reduce 128 of the 256 channels for the
// same 32-hw tile; partials combined through LDS. 576 blocks x 8 waves.
// Inner loop: running pointer (strength-reduced) + unconditional speculative
// prefetch 4 K-steps ahead so demand loads hit GL2.
// ---------------------------------------------------------------------------
#define TILES         2
#define HW_PER_WAVE   (TILES * 16)           // 32
#define TILE_GROUPS   4                      // hw sub-tiles per block
#define HW_PER_BLOCK  (TILE_GROUPS * HW_PER_WAVE)  // 128
#define BLK_PER_BATCH (HW_ / HW_PER_BLOCK)   // 72
#define CH_PER_WAVE   (C_ / 2)               // 128

__global__ __launch_bounds__(256) void corr_wmma_kernel(
    const float* __restrict__ fmap1, const float* __restrict__ m2,
    float* __restrict__ out) {
  const int batch = blockIdx.x / BLK_PER_BATCH;
  const int blk   = blockIdx.x % BLK_PER_BATCH;
  const int tid   = threadIdx.x;

  __shared__ float sm2[C_];
  __shared__ float part[8][HW_PER_WAVE];     // per-wave partial tiles

  sm2[tid] = m2[batch * C_ + tid];
  __syncthreads();

  const int wv    = tid >> 5;
  const int lane  = tid & 31;
  const int m     = lane & 15;     // M row (A) / N col (B,D)
  const int kg    = lane >> 4;     // lane group: K={0,1} vs K={2,3}
  const int tgrp  = wv & 3;        // which 32-hw sub-tile
  const int chalf = wv >> 2;       // which half of the channels
  const int hw0   = blk * HW_PER_BLOCK + tgrp * HW_PER_WAVE;
  const int cbase = chalf * CH_PER_WAVE;

  const float* f1 = fmap1 + (size_t)batch * C_ * HW_;

  v8f acc[TILES];
#pragma unroll
  for (int t = 0; t < TILES; ++t) acc[t] = (v8f){0.f,0.f,0.f,0.f,0.f,0.f,0.f,0.f};

  // Running induction variables -> backend strength-reduces to constant adds.
  const float* a0p = f1 + (size_t)(cbase + 2 * kg) * HW_ + hw0 + m;
  const float* sp  = &sm2[cbase + 2 * kg];

#pragma unroll 4
  for (int c0 = 0; c0 < CH_PER_WAVE; c0 += 4) {
    // B 4x16: VGPR0 -> K = 2*kg, VGPR1 -> K = 2*kg+1 (broadcast over N)
    v2f bm;
    bm[0] = sp[0];
    bm[1] = sp[1];
    // Speculative prefetch 4 K-steps (16 channels) ahead; OOB is dropped by HW.
    __builtin_prefetch(a0p + 16 * (size_t)HW_, 0, 1);
    // A 16x4: VGPR0 -> K = 2*kg, VGPR1 -> K = 2*kg+1; M = lane%16
    v2f am0, am1;
    am0[0] = a0p[0];
    am0[1] = a0p[HW_];
    am1[0] = a0p[16];
    am1[1] = a0p[HW_ + 16];
    acc[0] = __builtin_amdgcn_wmma_f32_16x16x4_f32(
        false, am0, false, bm, (short)0, acc[0], false, false);
    acc[1] = __builtin_amdgcn_wmma_f32_16x16x4_f32(
        false, am1, false, bm, (short)0, acc[1], false, false);
    a0p += 4 * (size_t)HW_;
    sp  += 4;
  }

  // D layout: VGPR r -> M=r (lanes 0-15, N=lane) / M=r+8 (lanes 16-31, N=lane-16).
  // All N columns identical; lanes with N==0 (lane 0 and lane 16) dump 8 rows each
  // into LDS as this wave's partial tile.
  if (m == 0) {
#pragma unroll
    for (int t = 0; t < TILES; ++t) {
      float* pp = &part[wv][t * 16 + kg * 8];
      float4 lo = make_float4(acc[t][0], acc[t][1], acc[t][2], acc[t][3]);
      float4 hi = make_float4(acc[t][4], acc[t][5], acc[t][6], acc[t][7]);
      *(float4*)(pp)     = lo;
      *(float4*)(pp + 4) = hi;
    }
  }
  __syncthreads();

  // Combine the two channel-halves and store: threads 0..127 cover 4 tiles x 32 hw.
  if (tid < TILE_GROUPS * HW_PER_WAVE) {
    const int g = tid >> 5;                  // tile group
    const int i = tid & 31;                  // hw within tile
    const float v = part[g][i] + part[g + 4][i];
    out[(size_t)batch * HW_ + blk * HW_PER_BLOCK + g * HW_PER_WAVE + i] = v;
  }
}

// ---------------------------------------------------------------------------
extern "C" void kernel_launch(void* const* d_in, const int* in_sizes, int n_in,
                              void* d_out, int out_size, void* d_ws, size_t ws_size,
                              hipStream_t stream) {
  const float* fmap1 = (const float*)d_in[0];
  const float* fmap2 = (const float*)d_in[1];
  float* out = (float*)d_out;
  float* m2  = (float*)d_ws;      // B_*C_ = 2048 floats = 8 KB scratch

  corr_mean_kernel<<<B_ * C_, 256, 0, stream>>>(fmap2, m2);
  corr_wmma_kernel<<<B_ * BLK_PER_BATCH, 256, 0, stream>>>(fmap1, m2, out);
}